// GCNEncoder_56427280335130
// MI455X (gfx1250) — compile-verified
//
#include <hip/hip_runtime.h>

#define N_U 100000
#define N_I 50000
#define NE  1600000
#define DIM 128

typedef __attribute__((ext_vector_type(2))) float v2f;
typedef __attribute__((ext_vector_type(8))) float v8f;

__device__ __forceinline__ void atomicAddF(float* p, float v) {
    __hip_atomic_fetch_add(p, v, __ATOMIC_RELAXED, __HIP_MEMORY_SCOPE_AGENT);
}

// ---------------- degree accumulation (one thread per edge) ----------------
__global__ void deg_kernel(const int* __restrict__ src, const int* __restrict__ dst,
                           float* __restrict__ degS, float* __restrict__ degD, int nE) {
    int e = blockIdx.x * blockDim.x + threadIdx.x;
    if (e < nE) {
        atomicAddF(&degS[src[e]], 1.0f);
        atomicAddF(&degD[dst[e]], 1.0f);
    }
}

// ---------------- deg -> 1/sqrt(deg) (0 if deg==0), in place ----------------
__global__ void invsqrt_kernel(float* __restrict__ a, int n) {
    int i = blockIdx.x * blockDim.x + threadIdx.x;
    if (i < n) {
        float d = a[i];
        a[i] = (d > 0.0f) ? rsqrtf(d) : 0.0f;
    }
}

// ---------------- H = X @ W via V_WMMA_F32_16X16X4_F32 ----------------
// One wave (32 threads) computes one 16x16 output tile.
// grid.x = nrows/16, grid.y = DIM/16 = 8, block = 32. EXEC is all-ones (no
// divergence; dims are exact multiples of 16) as WMMA requires.
__global__ void gemm_wmma_f32(const float* __restrict__ X, const float* __restrict__ W,
                              float* __restrict__ H) {
    const int m0   = blockIdx.x << 4;
    const int n0   = blockIdx.y << 4;
    const int lane = threadIdx.x;          // 0..31
    const int r15  = lane & 15;
    const int kb   = (lane >> 4) << 1;     // 0 for lanes 0-15, 2 for lanes 16-31

    const float* __restrict__ xrow = X + (size_t)(m0 + r15) * DIM;

    v8f c = {};
    #pragma unroll
    for (int k0 = 0; k0 < DIM; k0 += 4) {
        // A 16x4 (MxK): VGPR0 = K=kb, VGPR1 = K=kb+1, M = lane&15
        v2f a;
        a.x = xrow[k0 + kb];
        a.y = xrow[k0 + kb + 1];
        // B 4x16 (KxN): rows striped across lanes within a VGPR:
        // VGPR0: K=kb, N=lane&15 ; VGPR1: K=kb+1, N=lane&15
        v2f b;
        b.x = W[(size_t)(k0 + kb)     * DIM + n0 + r15];
        b.y = W[(size_t)(k0 + kb + 1) * DIM + n0 + r15];
        c = __builtin_amdgcn_wmma_f32_16x16x4_f32(
                /*neg_a=*/false, a, /*neg_b=*/false, b,
                /*c_mod=*/(short)0, c, /*reuse_a=*/false, /*reuse_b=*/false);
    }

    // C/D 16x16 f32 layout: VGPR r -> (M = (lane>=16?8:0)+r, N = lane&15)
    const int mbase = m0 + ((lane >> 4) << 3);
    #pragma unroll
    for (int r = 0; r < 8; ++r)
        H[(size_t)(mbase + r) * DIM + n0 + r15] = c[r];
}

// ---------------- edge scatter: acc[dst] += h[src] * invS[src]*invD[dst] ----
// One wave per edge; each lane moves one float4 (32 lanes * 16B = 512B row).
__global__ void scatter_kernel(const float* __restrict__ H,
                               const int* __restrict__ src, const int* __restrict__ dst,
                               const float* __restrict__ invS, const float* __restrict__ invD,
                               float* __restrict__ acc, int nE) {
    int gid  = blockIdx.x * blockDim.x + threadIdx.x;
    int e    = gid >> 5;
    int lane = gid & 31;
    if (e >= nE) return;

    int s = src[e];
    int d = dst[e];
    float nrm = invS[s] * invD[d];

    const float4 v = *((const float4*)(H + (size_t)s * DIM) + lane);
    float* op = acc + (size_t)d * DIM + lane * 4;
    atomicAddF(op + 0, v.x * nrm);
    atomicAddF(op + 1, v.y * nrm);
    atomicAddF(op + 2, v.z * nrm);
    atomicAddF(op + 3, v.w * nrm);
}

// ---------------- finalize: buf = [relu]( scale * (buf + b1 + b2?) ) --------
__global__ void finalize_kernel(float* __restrict__ buf,
                                const float* __restrict__ b1,
                                const float* __restrict__ b2,
                                float scale, int do_relu, int n) {
    int i = blockIdx.x * blockDim.x + threadIdx.x;
    if (i < n) {
        int c = i & (DIM - 1);
        float v = buf[i] + b1[c] + (b2 ? b2[c] : 0.0f);
        v *= scale;
        if (do_relu) v = fmaxf(v, 0.0f);
        buf[i] = v;
    }
}

// ---------------------------------------------------------------------------
extern "C" void kernel_launch(void* const* d_in, const int* in_sizes, int n_in,
                              void* d_out, int out_size, void* d_ws, size_t ws_size,
                              hipStream_t stream) {
    const float* x_user = (const float*)d_in[0];
    const float* x_item = (const float*)d_in[1];
    const int* f_src = (const int*)d_in[2];
    const int* f_dst = (const int*)d_in[3];
    const int* r_src = (const int*)d_in[4];
    const int* r_dst = (const int*)d_in[5];
    const int* v_src = (const int*)d_in[6];
    const int* v_dst = (const int*)d_in[7];
    const float* W1f = (const float*)d_in[8];
    const float* W1r = (const float*)d_in[9];
    const float* W1v = (const float*)d_in[10];
    const float* W2f = (const float*)d_in[11];
    const float* W2r = (const float*)d_in[12];
    const float* W2v = (const float*)d_in[13];
    const float* b1f = (const float*)d_in[14];
    const float* b1r = (const float*)d_in[15];
    const float* b1v = (const float*)d_in[16];
    const float* b2f = (const float*)d_in[17];
    const float* b2r = (const float*)d_in[18];
    const float* b2v = (const float*)d_in[19];

    // ---- workspace carve-up (floats) ----
    float* w = (float*)d_ws;
    float* inv_fs = w; w += N_U;   // follows src degree (users)
    float* inv_fd = w; w += N_U;   // follows dst degree (users)
    float* inv_rs = w; w += N_U;   // rates   src degree (users)
    float* inv_rd = w; w += N_I;   // rates   dst degree (items)
    float* inv_vs = w; w += N_I;   // rev     src degree (items)
    float* inv_vd = w; w += N_U;   // rev     dst degree (users)
    const int n_inv = 4 * N_U + 2 * N_I;
    float* h_user = w; w += (size_t)N_U * DIM;  // reused per relation/layer
    float* h_item = w; w += (size_t)N_I * DIM;
    float* u1     = w; w += (size_t)N_U * DIM;  // layer-1 user acc -> act
    float* it1    = w; w += (size_t)N_I * DIM;  // layer-1 item acc -> act

    float* out_u = (float*)d_out;                      // [N_U, 128]
    float* out_i = out_u + (size_t)N_U * DIM;          // [N_I, 128]

    const int TPB = 256;
    const int degBlocks  = (NE + TPB - 1) / TPB;
    const int scatBlocks = ((size_t)NE * 32 + TPB - 1) / TPB;
    dim3 gemmU((N_U / 16), DIM / 16);
    dim3 gemmI((N_I / 16), DIM / 16);

    // ---- degrees -> inv sqrt (edges identical for both layers) ----
    hipMemsetAsync(inv_fs, 0, (size_t)n_inv * sizeof(float), stream);
    deg_kernel<<<degBlocks, TPB, 0, stream>>>(f_src, f_dst, inv_fs, inv_fd, NE);
    deg_kernel<<<degBlocks, TPB, 0, stream>>>(r_src, r_dst, inv_rs, inv_rd, NE);
    deg_kernel<<<degBlocks, TPB, 0, stream>>>(v_src, v_dst, inv_vs, inv_vd, NE);
    invsqrt_kernel<<<(n_inv + TPB - 1) / TPB, TPB, 0, stream>>>(inv_fs, n_inv);

    // ---- layer 1 ----
    hipMemsetAsync(u1,  0, (size_t)N_U * DIM * sizeof(float), stream);
    hipMemsetAsync(it1, 0, (size_t)N_I * DIM * sizeof(float), stream);

    gemm_wmma_f32<<<gemmU, 32, 0, stream>>>(x_user, W1f, h_user);
    scatter_kernel<<<scatBlocks, TPB, 0, stream>>>(h_user, f_src, f_dst, inv_fs, inv_fd, u1, NE);
    gemm_wmma_f32<<<gemmI, 32, 0, stream>>>(x_item, W1v, h_item);
    scatter_kernel<<<scatBlocks, TPB, 0, stream>>>(h_item, v_src, v_dst, inv_vs, inv_vd, u1, NE);
    finalize_kernel<<<((size_t)N_U * DIM + TPB - 1) / TPB, TPB, 0, stream>>>(
        u1, b1f, b1v, 0.5f, 1, N_U * DIM);

    gemm_wmma_f32<<<gemmU, 32, 0, stream>>>(x_user, W1r, h_user);
    scatter_kernel<<<scatBlocks, TPB, 0, stream>>>(h_user, r_src, r_dst, inv_rs, inv_rd, it1, NE);
    finalize_kernel<<<((size_t)N_I * DIM + TPB - 1) / TPB, TPB, 0, stream>>>(
        it1, b1r, nullptr, 1.0f, 1, N_I * DIM);

    // ---- layer 2 (accumulate straight into d_out) ----
    hipMemsetAsync(d_out, 0, (size_t)(N_U + N_I) * DIM * sizeof(float), stream);

    gemm_wmma_f32<<<gemmU, 32, 0, stream>>>(u1, W2f, h_user);
    scatter_kernel<<<scatBlocks, TPB, 0, stream>>>(h_user, f_src, f_dst, inv_fs, inv_fd, out_u, NE);
    gemm_wmma_f32<<<gemmI, 32, 0, stream>>>(it1, W2v, h_item);
    scatter_kernel<<<scatBlocks, TPB, 0, stream>>>(h_item, v_src, v_dst, inv_vs, inv_vd, out_u, NE);
    finalize_kernel<<<((size_t)N_U * DIM + TPB - 1) / TPB, TPB, 0, stream>>>(
        out_u, b2f, b2v, 0.5f, 0, N_U * DIM);

    gemm_wmma_f32<<<gemmU, 32, 0, stream>>>(u1, W2r, h_user);
    scatter_kernel<<<scatBlocks, TPB, 0, stream>>>(h_user, r_src, r_dst, inv_rs, inv_rd, out_i, NE);
    finalize_kernel<<<((size_t)N_I * DIM + TPB - 1) / TPB, TPB, 0, stream>>>(
        out_i, b2r, nullptr, 1.0f, 0, N_I * DIM);
}